// MaskedHeteroGAT_23106924052984
// MI455X (gfx1250) — compile-verified
//
#include <hip/hip_runtime.h>
#include <math.h>

// ---------------------------------------------------------------------------
// MaskedHeteroGAT collapsed-form kernel for MI455X (gfx1250).
//
// Mathematical reduction: layer-2's source features are identically zero, so
// h2[t] == b2[t] broadcast to every node, hence
//   s[t]   = softmax(b2[t] @ Ws[t])      (one 64-vector p_t per edge type)
//   cross  = E/C,  sq = (N^2/C) * sum(p_t^2)
//   link_t = sqrt(E - 2E/C + (N^2/C)*sum(p_t^2)) / N^2
//   ent    = ( sum_t H(p_t) + H(uniform_C) ) / 7
// Only inputs b2 [6,128] and Ws [6,128,64] matter. ~100 KFLOP total; the
// ~8 GFLOP of GEMMs and GB-scale edge gathers in the reference are dead code.
//
// Implementation: 1 block, 6 wave32s; wave t computes z = b2[t] @ Ws[t]
// via V_WMMA_F32_16X16X4_F32 with b2[t] in row 0 of the 16x4 A tile.
// A layout (ISA 7.12.2, 32-bit A 16x4): lanes 0-15 = rows M, VGPR0=K0,
// VGPR1=K1; lanes 16-31 = rows M, VGPR0=K2, VGPR1=K3.  Only lanes 0 and 16
// carry row M=0 -> nonzero data (all lanes load, then cndmask to zero).
// B layout (mirror, 4x16): lanes 0-15: VGPR0=B[K0][n],VGPR1=B[K1][n];
// lanes 16-31: VGPR0=B[K2][n],VGPR1=B[K3][n].
// D layout: VGPR0 lanes 0-15 = row M=0 (our result), lanes 16-31 = row M=8
// which is exactly 0 here.
// ---------------------------------------------------------------------------

typedef float v2f __attribute__((ext_vector_type(2)));
typedef float v8f __attribute__((ext_vector_type(8)));

#define HD   128
#define CC   64
#define NTYP 6

__global__ __launch_bounds__(32 * NTYP)
void het_gat_collapsed(const float* __restrict__ b2,   // [6,128]
                       const float* __restrict__ Ws,   // [6,128,64]
                       float* __restrict__ out,        // [1]
                       int n_nodes, int n_edges)
{
    __shared__ float red_link[NTYP];
    __shared__ float red_ent[NTYP];

    const int t    = threadIdx.x >> 5;   // wave id == edge type
    const int lane = threadIdx.x & 31;
    const bool lo  = lane < 16;
    const int  l16 = lane & 15;

    const float* bt = b2 + t * HD;
    const float* wt = Ws + t * HD * CC;

    // ---- z[c] = sum_k b2[t][k] * Ws[t][k][c] via f32 WMMA -----------------
    v8f acc[4];
#pragma unroll
    for (int j = 0; j < 4; ++j)
        acc[j] = (v8f){0.f, 0.f, 0.f, 0.f, 0.f, 0.f, 0.f, 0.f};

    for (int k0 = 0; k0 < HD; k0 += 4) {
        const int krow = lo ? k0 : (k0 + 2);     // K-pair owned by this half-wave

        // A tile: every lane does one unconditional 8B load of its K-pair,
        // then zeroes it unless it is lane 0 / lane 16 (row M=0).
        const v2f apair = *(const v2f*)(bt + krow);   // 8B-aligned (krow even)
        v2f av;
        av.x = (l16 == 0) ? apair.x : 0.f;
        av.y = (l16 == 0) ? apair.y : 0.f;

#pragma unroll
        for (int j = 0; j < 4; ++j) {
            const int n0 = j * 16;
            v2f bv;
            bv.x = wt[(krow + 0) * CC + n0 + l16];
            bv.y = wt[(krow + 1) * CC + n0 + l16];
            // 8 args: (neg_a, A, neg_b, B, c_mod, C, reuse_a, reuse_b)
            acc[j] = __builtin_amdgcn_wmma_f32_16x16x4_f32(
                false, av, false, bv, (short)0, acc[j], false, false);
        }
    }

    // lanes 0-15: z[16j + lane] = acc[j][0]; lanes 16-31 hold exact zeros.
    float z0 = acc[0][0], z1 = acc[1][0], z2 = acc[2][0], z3 = acc[3][0];

    // ---- wave softmax over 64 values --------------------------------------
    float m = lo ? fmaxf(fmaxf(z0, z1), fmaxf(z2, z3)) : -3.0e38f;
#pragma unroll
    for (int off = 16; off; off >>= 1)
        m = fmaxf(m, __shfl_xor(m, off, 32));

    float e0 = lo ? __expf(z0 - m) : 0.f;
    float e1 = lo ? __expf(z1 - m) : 0.f;
    float e2 = lo ? __expf(z2 - m) : 0.f;
    float e3 = lo ? __expf(z3 - m) : 0.f;

    float S = e0 + e1 + e2 + e3;
#pragma unroll
    for (int off = 16; off; off >>= 1)
        S += __shfl_xor(S, off, 32);

    const float inv = 1.0f / S;
    const float p0 = e0 * inv, p1 = e1 * inv, p2 = e2 * inv, p3 = e3 * inv;

    float sq = p0 * p0 + p1 * p1 + p2 * p2 + p3 * p3;
    // entropy terms: for masked lanes p==0 -> contribute -0*log(1e-15)=0,
    // but keep the explicit predication to avoid -0*large rounding noise.
    float Hl = lo ? -(p0 * logf(p0 + 1e-15f) + p1 * logf(p1 + 1e-15f) +
                      p2 * logf(p2 + 1e-15f) + p3 * logf(p3 + 1e-15f))
                  : 0.f;
#pragma unroll
    for (int off = 16; off; off >>= 1) {
        sq += __shfl_xor(sq, off, 32);
        Hl += __shfl_xor(Hl, off, 32);
    }

    if (lane == 0) {
        const double E  = (double)n_edges;
        const double N2 = (double)n_nodes * (double)n_nodes;
        double inner = E - 2.0 * E / (double)CC + (N2 / (double)CC) * (double)sq;
        if (inner < 0.0) inner = 0.0;
        red_link[t] = (float)(sqrt(inner) / N2);
        red_ent[t]  = Hl;
    }
    __syncthreads();

    if (threadIdx.x == 0) {
        double link = 0.0, ent = 0.0;
#pragma unroll
        for (int i = 0; i < NTYP; ++i) {
            link += (double)red_link[i];
            ent  += (double)red_ent[i];
        }
        // uniform s_pkg rows: H = -C * (1/C) * log(1/C + 1e-15)
        const double hpkg = -log(1.0 / (double)CC + 1e-15);
        out[0] = (float)(link + (ent + hpkg) / 7.0);
    }
}

extern "C" void kernel_launch(void* const* d_in, const int* in_sizes, int n_in,
                              void* d_out, int out_size, void* d_ws, size_t ws_size,
                              hipStream_t stream) {
    // setup_inputs() order:
    //  0..6  : x_* node features      (unused: dead w.r.t. output)
    //  7..12 : ei_* edge indices      (unused: dead w.r.t. output)
    //  13..19: Wl1,Wr1,a1,b1,Wl2,Wr2,a2 (unused)
    //  20    : b2  [6,128]
    //  21    : Ws  [6,128,64]
    const float* b2 = (const float*)d_in[20];
    const float* Ws = (const float*)d_in[21];
    float* out = (float*)d_out;

    const int n_nodes = in_sizes[0] / HD;   // 20000
    const int n_edges = in_sizes[7] / 2;    // 200000

    het_gat_collapsed<<<dim3(1), dim3(32 * NTYP), 0, stream>>>(
        b2, Ws, out, n_nodes, n_edges);
}